// NeighborRoutingConv_58420145160706
// MI455X (gfx1250) — compile-verified
//
#include <hip/hip_runtime.h>
#include <math.h>

// ---------------- problem constants ----------------
#define N_NODES 50000
#define N_EDGES 800000
#define IN_DIM  256
#define OUT_DIM 256
#define KH      8      // heads
#define DK      32     // dims per head
#define NEG_SLOPE 0.2f

typedef float v2f __attribute__((ext_vector_type(2)));
typedef float v8f __attribute__((ext_vector_type(8)));

// ordered-uint mapping so atomicMax(u32) == float max (monotonic bijection)
__device__ __forceinline__ unsigned f2o(float f) {
    unsigned u = __float_as_uint(f);
    return (u & 0x80000000u) ? ~u : (u | 0x80000000u);
}
__device__ __forceinline__ float o2f(unsigned u) {
    return (u & 0x80000000u) ? __uint_as_float(u & 0x7FFFFFFFu) : __uint_as_float(~u);
}
#define ORD_NEG_INF 0x007FFFFFu   // f2o(-inf)

__device__ __forceinline__ float lrelu(float x) {
    return x > 0.0f ? x : NEG_SLOPE * x;
}

// ---------------- kernel 1: Wh = h @ W^T via V_WMMA_F32_16X16X4_F32 ----------------
// One wave computes a 16-row x 256-col stripe: 16 accumulator tiles (16 x v8f),
// K-loop in steps of 4. h is read exactly once; W (256KB) stays hot in L2.
__global__ __launch_bounds__(128)
void gemm_wmma_f32(const float* __restrict__ h, const float* __restrict__ W,
                   float* __restrict__ Wh) {
    const int lane = threadIdx.x & 31;
    const int wave = threadIdx.x >> 5;
    const int rowblk = blockIdx.x * 4 + wave;          // 16-row block index
    if (rowblk * 16 >= N_NODES) return;                // wave-uniform exit (EXEC stays all-1)
    const int m0   = rowblk * 16;
    const int half = lane >> 4;                        // 0: K0/K1 slot, 1: K2/K3 slot
    const int lm   = lane & 15;                        // row (A) / col (B) within tile

    v8f acc[16];
#pragma unroll
    for (int nt = 0; nt < 16; ++nt) acc[nt] = (v8f){0,0,0,0,0,0,0,0};

    for (int k0 = 0; k0 < IN_DIM; k0 += 4) {
        // A fragment: 16x4 tile of h.  lane<16 holds K=k0,k0+1; lane>=16 holds K=k0+2,k0+3
        const float2 av = *(const float2*)(h + (size_t)(m0 + lm) * IN_DIM + k0 + 2 * half);
        v2f afrag; afrag.x = av.x; afrag.y = av.y;
#pragma unroll
        for (int nt = 0; nt < 16; ++nt) {
            // B fragment: 4x16 tile of W^T  => B[k][n] = W[n][k]
            const float2 bv = *(const float2*)(W + (size_t)(nt * 16 + lm) * IN_DIM + k0 + 2 * half);
            v2f bfrag; bfrag.x = bv.x; bfrag.y = bv.y;
            acc[nt] = __builtin_amdgcn_wmma_f32_16x16x4_f32(
                false, afrag, false, bfrag, (short)0, acc[nt], false, false);
        }
    }

    // C/D layout: VGPR r -> row m0 + r + 8*half, col = nt*16 + lm
#pragma unroll
    for (int nt = 0; nt < 16; ++nt) {
#pragma unroll
        for (int r = 0; r < 8; ++r) {
            Wh[(size_t)(m0 + r + 8 * half) * OUT_DIM + nt * 16 + lm] = acc[nt][r];
        }
    }
}

// ---------------- kernel 2: a[n,k] = sum_d Wh[n,k*32+d]*attn[k*32+d] ----------------
// one wave per node, lane = d, shuffle-reduce across 32 lanes per head
__global__ __launch_bounds__(256)
void node_logits(const float* __restrict__ Wh, const float* __restrict__ attn,
                 float* __restrict__ a) {
    const int lane = threadIdx.x & 31;
    const int node = (blockIdx.x * (blockDim.x >> 5)) + (threadIdx.x >> 5);
    if (node >= N_NODES) return;
#pragma unroll
    for (int k = 0; k < KH; ++k) {
        float v = Wh[(size_t)node * OUT_DIM + k * DK + lane] * attn[k * DK + lane];
#pragma unroll
        for (int off = 16; off > 0; off >>= 1)
            v += __shfl_down(v, off, 32);
        if (lane == 0) a[(size_t)node * KH + k] = v;
    }
}

// ---------------- kernel 3: init out / e_sum / e_max ----------------
__global__ __launch_bounds__(256)
void init_bufs(float* __restrict__ out, float* __restrict__ esum,
               unsigned* __restrict__ emax) {
    const size_t i = (size_t)blockIdx.x * blockDim.x + threadIdx.x;
    if (i < (size_t)N_NODES * OUT_DIM) out[i] = 0.0f;
    if (i < (size_t)N_NODES * KH) { esum[i] = 0.0f; emax[i] = ORD_NEG_INF; }
}

// ---------------- kernel 4: segment max of edge logits ----------------
// one thread per (edge, head)
__global__ __launch_bounds__(256)
void edge_max(const int* __restrict__ src, const int* __restrict__ dst,
              const float* __restrict__ a, unsigned* __restrict__ emax) {
    const size_t i = (size_t)blockIdx.x * blockDim.x + threadIdx.x;
    if (i >= (size_t)N_EDGES * KH) return;
    const int e = (int)(i >> 3);
    const int k = (int)(i & 7);
    const int s = src[e], d = dst[e];
    const float v = lrelu(a[(size_t)s * KH + k] + a[(size_t)d * KH + k]);
    atomicMax(&emax[(size_t)d * KH + k], f2o(v));
}

// ---------------- kernel 5: segment sum of exp(e - max) ----------------
__global__ __launch_bounds__(256)
void edge_expsum(const int* __restrict__ src, const int* __restrict__ dst,
                 const float* __restrict__ a, const unsigned* __restrict__ emax,
                 float* __restrict__ esum) {
    const size_t i = (size_t)blockIdx.x * blockDim.x + threadIdx.x;
    if (i >= (size_t)N_EDGES * KH) return;
    const int e = (int)(i >> 3);
    const int k = (int)(i & 7);
    const int s = src[e], d = dst[e];
    const float v = lrelu(a[(size_t)s * KH + k] + a[(size_t)d * KH + k]);
    const float m = o2f(emax[(size_t)d * KH + k]);
    atomicAdd(&esum[(size_t)d * KH + k], __expf(v - m));
}

// ---------------- kernel 6: message passing ----------------
// one wave per edge; iteration j covers head k=j (wave-uniform softmax math),
// lanes do a coalesced 128B gather of Wh[src] and coalesced f32 atomic adds to out[dst]
__global__ __launch_bounds__(256)
void edge_message(const int* __restrict__ src, const int* __restrict__ dst,
                  const float* __restrict__ a, const unsigned* __restrict__ emax,
                  const float* __restrict__ esum, const float* __restrict__ Wh,
                  float* __restrict__ out) {
    const int lane = threadIdx.x & 31;
    const int e = blockIdx.x * (blockDim.x >> 5) + (threadIdx.x >> 5);
    if (e >= N_EDGES) return;
    const int s = src[e], d = dst[e];
    const float* whs = Wh + (size_t)s * OUT_DIM;
    float* od = out + (size_t)d * OUT_DIM;
#pragma unroll
    for (int j = 0; j < KH; ++j) {
        const float logit = lrelu(a[(size_t)s * KH + j] + a[(size_t)d * KH + j]);
        const float m  = o2f(emax[(size_t)d * KH + j]);
        const float su = esum[(size_t)d * KH + j];
        const float alpha = __expf(logit - m) / su;
        const int idx = j * DK + lane;
        atomicAdd(&od[idx], whs[idx] * alpha);
    }
}

// ---------------- host launcher ----------------
extern "C" void kernel_launch(void* const* d_in, const int* in_sizes, int n_in,
                              void* d_out, int out_size, void* d_ws, size_t ws_size,
                              hipStream_t stream) {
    const float* h    = (const float*)d_in[0];
    const int*   esrc = (const int*)d_in[1];
    const int*   edst = (const int*)d_in[2];
    const float* W    = (const float*)d_in[3];
    const float* attn = (const float*)d_in[4];
    float* out = (float*)d_out;

    // workspace layout (all fp32 / u32), ~56 MB total
    float*    Wh   = (float*)d_ws;                            // N*256
    float*    a    = Wh + (size_t)N_NODES * OUT_DIM;          // N*8
    unsigned* emax = (unsigned*)(a + (size_t)N_NODES * KH);   // N*8
    float*    esum = (float*)(emax + (size_t)N_NODES * KH);   // N*8

    // 1) GEMM: 3125 row-blocks of 16, 4 waves/block
    {
        const int rowblks = N_NODES / 16;                     // 3125
        const int blocks = (rowblks + 3) / 4;                 // 782 (last has uniform-exit waves)
        gemm_wmma_f32<<<blocks, 128, 0, stream>>>(h, W, Wh);
    }
    // 2) per-node head logits: 1 wave per node, 8 waves/block
    {
        const int blocks = (N_NODES + 7) / 8;                 // 6250
        node_logits<<<blocks, 256, 0, stream>>>(Wh, attn, a);
    }
    // 3) init
    {
        const int total = N_NODES * OUT_DIM;
        init_bufs<<<(total + 255) / 256, 256, 0, stream>>>(out, esum, emax);
    }
    // 4) segment max
    {
        const int total = N_EDGES * KH;                       // 6.4M
        edge_max<<<(total + 255) / 256, 256, 0, stream>>>(esrc, edst, a, emax);
    }
    // 5) segment sum of exponentials
    {
        const int total = N_EDGES * KH;
        edge_expsum<<<(total + 255) / 256, 256, 0, stream>>>(esrc, edst, a, emax, esum);
    }
    // 6) messages: 1 wave per edge, 8 waves/block
    {
        const int blocks = (N_EDGES + 7) / 8;                 // 100000
        edge_message<<<blocks, 256, 0, stream>>>(esrc, edst, a, emax, esum, Wh, out);
    }
}